// MSDBlock_69380901699704
// MI455X (gfx1250) — compile-verified
//
#include <hip/hip_runtime.h>

typedef float v2f __attribute__((ext_vector_type(2)));
typedef float v8f __attribute__((ext_vector_type(8)));

#define BATCH 4
#define HH 512
#define WW 512
#define NCH_TOTAL 25   // 1 + 2*12
#define DEPTH 12
#define KMAX4 208      // ceil(23*9 / 4) * 4

__device__ __forceinline__ int reflect_idx(int t, int n) {
    // single reflection suffices: |offset| = dil <= 12 << 512
    t = (t < 0) ? -t : t;
    t = (t >= n) ? (2 * n - 2 - t) : t;
    return t;
}

// Copy x (B,1,H,W) into channel 0 of the (B,25,H,W) output/concat buffer.
__global__ __launch_bounds__(256) void msd_copy_input(const float4* __restrict__ x,
                                                      float4* __restrict__ out) {
    const int perB = HH * WW / 4;
    int idx = blockIdx.x * blockDim.x + threadIdx.x;
    if (idx >= BATCH * perB) return;
    int b = idx / perB;
    int r = idx - b * perB;
    out[(size_t)b * NCH_TOTAL * perB + r] = x[idx];
}

// K-loop for two output rows, 4 pixel groups each (8 WMMAs / chunk).
// INTERIOR: x + dx*dil provably in [0, W) -> base pointer + immediate offsets.
template <bool INTERIOR>
static __device__ __forceinline__ void
msd_kloop(const float* __restrict__ inb, const float* __restrict__ wlds,
          const int2* __restrict__ meta0, const int2* __restrict__ meta1,
          int Ktot, int roff, int khalf, int x0,
          v8f& c00, v8f& c01, v8f& c02, v8f& c03,
          v8f& c10, v8f& c11, v8f& c12, v8f& c13) {
    for (int kk = 0; kk < Ktot; kk += 4) {
        const int k0 = kk + khalf;
        v2f a = *(const v2f*)&wlds[roff + k0];           // ds_load_b64 (8B aligned)
        int4 M0 = *(const int4*)&meta0[k0];              // ds_load_b128: {rb,dxd} x2 (row y0)
        int4 M1 = *(const int4*)&meta1[k0];              // ds_load_b128 (row y1)

        v2f b00, b01, b02, b03, b10, b11, b12, b13;
        if (INTERIOR) {
            const float* q00 = inb + (unsigned)(M0.x + x0 + M0.y);
            const float* q01 = inb + (unsigned)(M0.z + x0 + M0.w);
            const float* q10 = inb + (unsigned)(M1.x + x0 + M1.y);
            const float* q11 = inb + (unsigned)(M1.z + x0 + M1.w);
            b00.x = q00[0];  b00.y = q01[0];
            b01.x = q00[16]; b01.y = q01[16];
            b02.x = q00[32]; b02.y = q01[32];
            b03.x = q00[48]; b03.y = q01[48];
            b10.x = q10[0];  b10.y = q11[0];
            b11.x = q10[16]; b11.y = q11[16];
            b12.x = q10[32]; b12.y = q11[32];
            b13.x = q10[48]; b13.y = q11[48];
        } else {
            const float* q00 = inb + (unsigned)M0.x;
            const float* q01 = inb + (unsigned)M0.z;
            const float* q10 = inb + (unsigned)M1.x;
            const float* q11 = inb + (unsigned)M1.z;
            const int t00 = x0 + M0.y, t01 = x0 + M0.w;
            const int t10 = x0 + M1.y, t11 = x0 + M1.w;
            b00.x = q00[reflect_idx(t00,      WW)]; b00.y = q01[reflect_idx(t01,      WW)];
            b01.x = q00[reflect_idx(t00 + 16, WW)]; b01.y = q01[reflect_idx(t01 + 16, WW)];
            b02.x = q00[reflect_idx(t00 + 32, WW)]; b02.y = q01[reflect_idx(t01 + 32, WW)];
            b03.x = q00[reflect_idx(t00 + 48, WW)]; b03.y = q01[reflect_idx(t01 + 48, WW)];
            b10.x = q10[reflect_idx(t10,      WW)]; b10.y = q11[reflect_idx(t11,      WW)];
            b11.x = q10[reflect_idx(t10 + 16, WW)]; b11.y = q11[reflect_idx(t11 + 16, WW)];
            b12.x = q10[reflect_idx(t10 + 32, WW)]; b12.y = q11[reflect_idx(t11 + 32, WW)];
            b13.x = q10[reflect_idx(t10 + 48, WW)]; b13.y = q11[reflect_idx(t11 + 48, WW)];
        }
        // A fragment reused by all 8 WMMAs in the chunk.
        c00 = __builtin_amdgcn_wmma_f32_16x16x4_f32(false, a, false, b00, (short)0, c00, false, false);
        c01 = __builtin_amdgcn_wmma_f32_16x16x4_f32(false, a, false, b01, (short)0, c01, false, false);
        c02 = __builtin_amdgcn_wmma_f32_16x16x4_f32(false, a, false, b02, (short)0, c02, false, false);
        c03 = __builtin_amdgcn_wmma_f32_16x16x4_f32(false, a, false, b03, (short)0, c03, false, false);
        c10 = __builtin_amdgcn_wmma_f32_16x16x4_f32(false, a, false, b10, (short)0, c10, false, false);
        c11 = __builtin_amdgcn_wmma_f32_16x16x4_f32(false, a, false, b11, (short)0, c11, false, false);
        c12 = __builtin_amdgcn_wmma_f32_16x16x4_f32(false, a, false, b12, (short)0, c12, false, false);
        c13 = __builtin_amdgcn_wmma_f32_16x16x4_f32(false, a, false, b13, (short)0, c13, false, false);
    }
}

// One MSD layer as implicit GEMM on V_WMMA_F32_16X16X4_F32.
//   A (16xK): weights in LDS (rows 0..1 real, row "2" zero pad, K padded to 4).
//   B (Kx16): im2col gather; per-k {rowBase, dx*dil} metadata in LDS, per row.
//   D rows 0,1 = the two output channels; 2 rows x 4 groups = 128 px per wave.
// Grid = B*H/2 (two image rows per 256-thread workgroup).
__global__ __launch_bounds__(256) void msd_layer_wmma(const float* __restrict__ bufin,
                                                      float* __restrict__ bufout,
                                                      const float* __restrict__ w,
                                                      const float* __restrict__ bias,
                                                      int layer, int nin, int dil) {
    const int Ktot  = nin * 9;
    const int Ktot4 = (Ktot + 3) & ~3;

    __shared__ alignas(16) float wlds[3 * KMAX4];   // rows 0,1 = weights; row 2 = zeros
    __shared__ alignas(16) int2 meta[2][KMAX4];     // per row: {rowBaseIdx, dx*dil} per k

    const int tid = threadIdx.x;
    const int b   = blockIdx.x >> 8;                // blockIdx.x = b*(H/2) + yr
    const int y0  = (blockIdx.x & 255) * 2;
    const int y1  = y0 + 1;

    // ---- stage weights (zero-padded) ----
    for (int t = tid; t < 3 * Ktot4; t += 256) {
        int r = t / Ktot4;
        int k = t - r * Ktot4;
        wlds[t] = (r < 2 && k < Ktot) ? w[r * Ktot + k] : 0.0f;
    }
    // ---- stage per-k metadata for both rows ----
    for (int k = tid; k < Ktot4; k += 256) {
        int rb0 = 0, rb1 = 0, dxd = 0;
        if (k < Ktot) {
            int cin = k / 9, tap = k - cin * 9;
            int dy  = tap / 3 - 1, dx = tap % 3 - 1;
            rb0 = (cin * HH + reflect_idx(y0 + dy * dil, HH)) * WW;
            rb1 = (cin * HH + reflect_idx(y1 + dy * dil, HH)) * WW;
            dxd = dx * dil;
        }
        meta[0][k] = make_int2(rb0, dxd);
        meta[1][k] = make_int2(rb1, dxd);
    }
    __syncthreads();

    const int wave  = tid >> 5;
    const int lane  = tid & 31;
    const int m     = lane & 15;               // M row (A) / N col (B)
    const int khalf = (lane >> 4) << 1;        // lanes 0-15 -> K 0,1; lanes 16-31 -> K 2,3
    const int xbase = wave * 64;
    const int x0    = xbase + m;
    const int roff  = (m < 2 ? m : 2) * Ktot4; // A row offset (row 2 = zeros)

    const float* inb = bufin + (size_t)b * NCH_TOTAL * HH * WW;

    v8f c00 = {}, c01 = {}, c02 = {}, c03 = {};
    v8f c10 = {}, c11 = {}, c12 = {}, c13 = {};

    // wave-uniform interior test (x + dx*dil in [0, W) for the whole 64-px tile)
    const int xb_u = __builtin_amdgcn_readfirstlane(xbase);
    if (xb_u >= 16 && xb_u <= 432) {
        msd_kloop<true>(inb, wlds, meta[0], meta[1], Ktot, roff, khalf, x0,
                        c00, c01, c02, c03, c10, c11, c12, c13);
    } else {
        msd_kloop<false>(inb, wlds, meta[0], meta[1], Ktot, roff, khalf, x0,
                         c00, c01, c02, c03, c10, c11, c12, c13);
    }

    // ---- epilogue: lanes 0-15 hold M=0 in c[0], M=1 in c[1] (ISA C/D layout) ----
    const float bv0 = bias[2 * layer];
    const float bv1 = bias[2 * layer + 1];
    if (lane < 16) {
        const int co = 1 + 2 * layer;
        float* o00 = bufout + (((size_t)b * NCH_TOTAL + co) * HH + y0) * WW;
        float* o01 = o00 + (size_t)HH * WW;      // channel co+1, row y0
        float* o10 = o00 + WW;                   // channel co,   row y1
        float* o11 = o01 + WW;                   // channel co+1, row y1
        float r;
        r = c00[0] + bv0; o00[x0]      = r > 0.0f ? r : 0.0f;
        r = c00[1] + bv1; o01[x0]      = r > 0.0f ? r : 0.0f;
        r = c01[0] + bv0; o00[x0 + 16] = r > 0.0f ? r : 0.0f;
        r = c01[1] + bv1; o01[x0 + 16] = r > 0.0f ? r : 0.0f;
        r = c02[0] + bv0; o00[x0 + 32] = r > 0.0f ? r : 0.0f;
        r = c02[1] + bv1; o01[x0 + 32] = r > 0.0f ? r : 0.0f;
        r = c03[0] + bv0; o00[x0 + 48] = r > 0.0f ? r : 0.0f;
        r = c03[1] + bv1; o01[x0 + 48] = r > 0.0f ? r : 0.0f;
        r = c10[0] + bv0; o10[x0]      = r > 0.0f ? r : 0.0f;
        r = c10[1] + bv1; o11[x0]      = r > 0.0f ? r : 0.0f;
        r = c11[0] + bv0; o10[x0 + 16] = r > 0.0f ? r : 0.0f;
        r = c11[1] + bv1; o11[x0 + 16] = r > 0.0f ? r : 0.0f;
        r = c12[0] + bv0; o10[x0 + 32] = r > 0.0f ? r : 0.0f;
        r = c12[1] + bv1; o11[x0 + 32] = r > 0.0f ? r : 0.0f;
        r = c13[0] + bv0; o10[x0 + 48] = r > 0.0f ? r : 0.0f;
        r = c13[1] + bv1; o11[x0 + 48] = r > 0.0f ? r : 0.0f;
    }
}

extern "C" void kernel_launch(void* const* d_in, const int* in_sizes, int n_in,
                              void* d_out, int out_size, void* d_ws, size_t ws_size,
                              hipStream_t stream) {
    (void)in_sizes; (void)n_in; (void)out_size; (void)d_ws; (void)ws_size;

    const float* x    = (const float*)d_in[0];
    const float* bias = (const float*)d_in[1];
    float* out        = (float*)d_out;

    // Stage 0: x -> channel 0 of the concat buffer (d_out doubles as workspace;
    // the whole 100 MB tensor stays resident in the 192 MB L2).
    const int total4 = BATCH * HH * WW / 4;
    msd_copy_input<<<(total4 + 255) / 256, 256, 0, stream>>>((const float4*)x, (float4*)out);

    // 12 dependent layers: layer i reads channels [0, 1+2i), writes 1+2i and 2+2i.
    for (int i = 0; i < DEPTH; ++i) {
        const int nin = 1 + 2 * i;
        const float* wp = (const float*)d_in[2 + i];
        msd_layer_wmma<<<BATCH * HH / 2, 256, 0, stream>>>(out, out, wp, bias, i, nin, i + 1);
    }
}